// Entmax15_43602507989420
// MI455X (gfx1250) — compile-verified
//
#include <hip/hip_runtime.h>
#include <cstddef>
#include <cstdint>

// ---------------------------------------------------------------------------
// entmax-1.5 bisection (Peters & Martins) forward, rows of D=1024 fp32.
// One wave32 per row; row lives in 32 VGPRs/lane (8 x float4).
// 50 bisection iterations evaluate f(tau)=sum relu(x*(a-1)-tau)^(1/(a-1))-1
// entirely out of registers; cross-lane sums via 5-step xor butterfly.
// Data staged Global->LDS with gfx1250 async-to-LDS ops (ASYNCcnt-tracked).
// 4 independent FMA accumulators break the dependence chain for VOPD issue.
// ---------------------------------------------------------------------------

#define ENTMAX_D 1024
#define ROWS_PER_BLOCK 8     // 8 waves * 32 lanes = 256 threads
#define CHUNKS 8             // 8 float4 per lane * 32 lanes * 4 = 1024 floats

typedef float fvec4 __attribute__((ext_vector_type(4)));
typedef int v4i __attribute__((vector_size(16)));   // async builtin's pointee

#if defined(__HIP_DEVICE_COMPILE__) && \
    __has_builtin(__builtin_amdgcn_global_load_async_to_lds_b128)
#define ENTMAX_ASYNC_LDS 1
typedef __attribute__((address_space(1))) v4i* gv4i_p;  // global (AS1)
typedef __attribute__((address_space(3))) v4i* lv4i_p;  // LDS (AS3)
#else
#define ENTMAX_ASYNC_LDS 0
#endif

__device__ __forceinline__ float wave_sum32(float v) {
#pragma unroll
  for (int m = 16; m >= 1; m >>= 1) v += __shfl_xor(v, m, 32);
  return v;
}

__device__ __forceinline__ float wave_max32(float v) {
#pragma unroll
  for (int m = 16; m >= 1; m >>= 1) v = fmaxf(v, __shfl_xor(v, m, 32));
  return v;
}

// p(t) = clip(t,0)^(1/(alpha-1)); SQ=true specializes alpha=1.5 (square).
template <bool SQ>
__device__ __forceinline__ float pfun(float t, float pinv) {
  t = fmaxf(t, 0.0f);
  if (SQ) return t * t;
  return (t > 0.0f) ? exp2f(pinv * log2f(t)) : 0.0f;
}

// f(tau)+1 = sum over row of p(x*am1 - tau); full-row sum broadcast to lanes.
// 4 independent accumulators: dependent FMA chain is 8 deep, not 32.
template <bool SQ>
__device__ __forceinline__ float fsum(const fvec4 (&xv)[CHUNKS], float am1,
                                      float tau, float pinv) {
  float a0 = 0.0f, a1 = 0.0f, a2 = 0.0f, a3 = 0.0f;
#pragma unroll
  for (int c = 0; c < CHUNKS; ++c) {
    // 3 VALU/elem in SQ path: v_fma (fold am1), v_max, v_fma (square+acc)
    float t0 = fmaxf(fmaf(xv[c].x, am1, -tau), 0.0f);
    float t1 = fmaxf(fmaf(xv[c].y, am1, -tau), 0.0f);
    float t2 = fmaxf(fmaf(xv[c].z, am1, -tau), 0.0f);
    float t3 = fmaxf(fmaf(xv[c].w, am1, -tau), 0.0f);
    if (SQ) {
      a0 = fmaf(t0, t0, a0);
      a1 = fmaf(t1, t1, a1);
      a2 = fmaf(t2, t2, a2);
      a3 = fmaf(t3, t3, a3);
    } else {
      a0 += (t0 > 0.0f) ? exp2f(pinv * log2f(t0)) : 0.0f;
      a1 += (t1 > 0.0f) ? exp2f(pinv * log2f(t1)) : 0.0f;
      a2 += (t2 > 0.0f) ? exp2f(pinv * log2f(t2)) : 0.0f;
      a3 += (t3 > 0.0f) ? exp2f(pinv * log2f(t3)) : 0.0f;
    }
  }
  return wave_sum32((a0 + a1) + (a2 + a3));
}

template <bool SQ>
__device__ __forceinline__ void solve_row(const fvec4 (&xv)[CHUNKS], float am1,
                                          float pinv, float* __restrict__ orow) {
  // max over row of Xs = x*am1 ; am1 >= 0 so max commutes with the scale
  float m0 = -3.402823466e+38f, m1 = -3.402823466e+38f;
#pragma unroll
  for (int c = 0; c < CHUNKS; ++c) {
    m0 = fmaxf(m0, fmaxf(xv[c].x, xv[c].y));
    m1 = fmaxf(m1, fmaxf(xv[c].z, xv[c].w));
  }
  const float mxs = wave_max32(fmaxf(m0, m1)) * am1;

  float tau_lo = mxs - 1.0f;                         // _gp(1, alpha) = 1
  const float tau_hi = mxs - exp2f(-10.0f * am1);    // (1/1024)^(alpha-1)
  const float f_lo = fsum<SQ>(xv, am1, tau_lo, pinv) - 1.0f;

  float dm = tau_hi - tau_lo;
  float tau_m = tau_lo;
  float sum_p = 0.0f;
  for (int it = 0; it < 50; ++it) {   // N_ITER = 50, faithful to reference
    dm *= 0.5f;
    tau_m = tau_lo + dm;
    sum_p = fsum<SQ>(xv, am1, tau_m, pinv);
    const float f_m = sum_p - 1.0f;
    tau_lo = (f_m * f_lo >= 0.0f) ? tau_m : tau_lo;  // uniform across lanes
  }

  // reference returns p_m (last iteration's tau_m) normalized by its sum
  const float inv = 1.0f / sum_p;
#pragma unroll
  for (int c = 0; c < CHUNKS; ++c) {
    fvec4 p;
    p.x = pfun<SQ>(fmaf(xv[c].x, am1, -tau_m), pinv) * inv;
    p.y = pfun<SQ>(fmaf(xv[c].y, am1, -tau_m), pinv) * inv;
    p.z = pfun<SQ>(fmaf(xv[c].z, am1, -tau_m), pinv) * inv;
    p.w = pfun<SQ>(fmaf(xv[c].w, am1, -tau_m), pinv) * inv;
    // write-once 512MB stream: keep it out of L2 with a nontemporal store
    __builtin_nontemporal_store(p, (fvec4*)(orow + c * 128));
  }
}

__global__ __launch_bounds__(256) void entmax_bisect_kernel(
    const float* __restrict__ x, const int* __restrict__ step_ptr,
    float* __restrict__ out, int rows) {
  const int lane = threadIdx.x & 31;
  const int wid = threadIdx.x >> 5;
  const int row = blockIdx.x * ROWS_PER_BLOCK + wid;

#if ENTMAX_ASYNC_LDS
  __shared__ __align__(16) float smem[ROWS_PER_BLOCK][ENTMAX_D];
#endif
  if (row >= rows) return;  // wave-uniform exit

  const size_t base = (size_t)row * ENTMAX_D + (size_t)(lane * 4);
  fvec4 xv[CHUNKS];

#if ENTMAX_ASYNC_LDS
  {
    const float* g = x + base;
    float* l = &smem[wid][lane * 4];
#pragma unroll
    for (int c = 0; c < CHUNKS; ++c) {
      // gfx1250 async copy Global->LDS, 16B/lane, tracked by ASYNCcnt
      __builtin_amdgcn_global_load_async_to_lds_b128(
          (gv4i_p)(g + c * 128), (lv4i_p)(l + c * 128),
          /*offset=*/0, /*cpol=*/0);
    }
#if __has_builtin(__builtin_amdgcn_s_wait_asynccnt)
    __builtin_amdgcn_s_wait_asynccnt(0);
#else
    asm volatile("s_wait_asynccnt 0" ::: "memory");
#endif
    asm volatile("" ::: "memory");  // order LDS reads after the wait
#pragma unroll
    for (int c = 0; c < CHUNKS; ++c)
      xv[c] = *(const fvec4*)(l + c * 128);  // ds_load_b128, row now in VGPRs
  }
#else
  {
    const fvec4* g4 = (const fvec4*)(x + (size_t)row * ENTMAX_D);
#pragma unroll
    for (int c = 0; c < CHUNKS; ++c) xv[c] = g4[c * 32 + lane];
  }
#endif

  // alpha schedule: t = min(step, 10000)/10000 ; alpha = 1 + 0.5*t
  const float t01 = fminf((float)(*step_ptr), 10000.0f) * (1.0f / 10000.0f);
  const float am1 = 0.5f * t01;      // alpha - 1
  const float pinv = 1.0f / am1;     // 1/(alpha-1)

  float* orow = out + base;
  if (pinv == 2.0f) {
    solve_row<true>(xv, am1, 2.0f, orow);   // alpha = 1.5 fast path (square)
  } else {
    solve_row<false>(xv, am1, pinv, orow);  // generic pow path
  }
}

extern "C" void kernel_launch(void* const* d_in, const int* in_sizes, int n_in,
                              void* d_out, int out_size, void* d_ws,
                              size_t ws_size, hipStream_t stream) {
  const float* x = (const float*)d_in[0];
  const int* step = (const int*)d_in[1];
  float* out = (float*)d_out;

  const int rows = in_sizes[0] / ENTMAX_D;  // (8*16*1024*1024)/1024 = 131072
  const int blocks = (rows + ROWS_PER_BLOCK - 1) / ROWS_PER_BLOCK;

  hipLaunchKernelGGL(entmax_bisect_kernel, dim3(blocks), dim3(256), 0, stream,
                     x, step, out, rows);
}